// FramePredictionBase_30210799960617
// MI455X (gfx1250) — compile-verified
//
#include <hip/hip_runtime.h>
#include <hip/hip_bf16.h>

// ---------------------------------------------------------------------------
// FramePrediction loss on MI455X (gfx1250), fp32 WMMA path.
//   B=32, SEQ=50, OBS=1024, HID=256
// Pipeline:
//   1) normalize rows of U_param/V_param
//   2) Q_U, Q_V = product of 256 Householder reflections (Q in LDS, 256KB)
//   3) W = S .* V^T ;  A = Q_U @ W               (WMMA f32 16x16x4, NT=4)
//   4) O[0]=C ; O[t]=O[t-1]@A, t=1..49           (WMMA, A async-staged in LDS)
//   5) G = O^T O ; By = O^T Y^T                  (WMMA, split-K=25, partials)
//   6) Gauss-Jordan solve G x0 = By              (single WGP, LDS 288KB)
//   7) Yhat=O@x0 fused residual SSE (per-block partials) + logdet, finalize
// All reductions fixed-order -> bitwise deterministic across replays.
// ---------------------------------------------------------------------------

typedef __attribute__((ext_vector_type(2))) float v2f;
typedef __attribute__((ext_vector_type(8))) float v8f;
typedef __attribute__((ext_vector_type(4))) int   v4i;

#define HID    256
#define OBS    1024
#define BATCH  32
#define SEQ    50
#define SO     (SEQ * OBS)     /* 51200 rows of O */
#define KSPLIT 25
#define KCHUNK (SO / KSPLIT)   /* 2048 */
#define RES_BLOCKS 400         /* (SO/16) wave-groups / 8 waves per block */

// gfx1250 async global->LDS copy (ASYNCcnt path), guarded so the file
// compiles on toolchains where the builtin is absent or renamed.
// Probe (round 3) showed the builtin takes pointers to int4 vectors:
//   arg0: AS(1) v4i*  (global src), arg1: AS(3) v4i* (LDS dst), i32, i32.
#if defined(__AMDGCN__) && \
    __has_builtin(__builtin_amdgcn_global_load_async_to_lds_b128) && \
    __has_builtin(__builtin_amdgcn_s_wait_asynccnt)
#define HAVE_ASYNC_LDS 1
typedef __attribute__((address_space(1))) v4i as1_v4i;
typedef __attribute__((address_space(3))) v4i as3_v4i;
#else
#define HAVE_ASYNC_LDS 0
#endif

// Cooperative 256-thread copy (multiple of 256*16 bytes) global -> LDS.
// Async path issues global_load_async_to_lds_b128 per 16B chunk and waits on
// ASYNCcnt; fallback round-trips through VGPRs as float4.
__device__ __forceinline__ void stage_to_lds_256(const float* __restrict__ g,
                                                 float* __restrict__ l,
                                                 int nfloat, int tid) {
#if HAVE_ASYNC_LDS
  const char* gs = (const char*)g;
  char* ls = (char*)l;
  int nchunk = nfloat * 4 / 16;
  for (int e = tid; e < nchunk; e += 256) {
    __builtin_amdgcn_global_load_async_to_lds_b128(
        (as1_v4i*)(gs + (size_t)e * 16), (as3_v4i*)(ls + (size_t)e * 16),
        0, 0);
  }
  __builtin_amdgcn_s_wait_asynccnt(0);
#else
  const float4* s4 = (const float4*)g;
  float4* d4 = (float4*)l;
  for (int e = tid; e < nfloat / 4; e += 256) d4[e] = s4[e];
#endif
  __syncthreads();
}

// ---- workspace layout (float offsets) -------------------------------------
#define WS_UNU   0
#define WS_UNV   (WS_UNU + HID*HID)
#define WS_QU    (WS_UNV + HID*HID)
#define WS_QV    (WS_QU  + HID*HID)
#define WS_W     (WS_QV  + HID*HID)
#define WS_A     (WS_W   + HID*HID)
#define WS_G     (WS_A   + HID*HID)
#define WS_BY    (WS_G   + HID*HID)
#define WS_X0    (WS_BY  + HID*BATCH)
#define WS_RES   (WS_X0  + HID*BATCH)            /* RES_BLOCKS partials */
#define WS_GP    (WS_RES + RES_BLOCKS)           /* KSPLIT*HID*HID      */
#define WS_BYP   (WS_GP  + KSPLIT*HID*HID)       /* KSPLIT*HID*BATCH    */
#define WS_O     (WS_BYP + KSPLIT*HID*BATCH)     /* 50*1024*256         */

// ---- row-normalize Householder vectors ------------------------------------
__global__ void normalize_rows_kernel(const float* __restrict__ Up,
                                      const float* __restrict__ Vp,
                                      float* __restrict__ ws) {
  __shared__ float red[HID];
  const float* src = (blockIdx.y == 0) ? Up : Vp;
  float* dst = ws + ((blockIdx.y == 0) ? WS_UNU : WS_UNV);
  int r = blockIdx.x, j = threadIdx.x;
  float v = src[r * HID + j];
  red[j] = v * v;
  __syncthreads();
  for (int s = HID / 2; s > 0; s >>= 1) {
    if (j < s) red[j] += red[j + s];
    __syncthreads();
  }
  float inv = rsqrtf(red[0]);
  dst[r * HID + j] = v * inv;
}

// ---- sequential Householder product, Q resident in LDS --------------------
// grid 2 (block 0 -> U, block 1 -> V), block 1024. dyn LDS 268,288 B.
__global__ void householder_chain_kernel(float* __restrict__ ws) {
  extern __shared__ float lds[];
  float* Q  = lds;                       // 65536
  float* u  = lds + HID * HID;           // 256
  float* t  = u + HID;                   // 256
  float* t4 = t + HID;                   // 1024
  const float* Un = ws + ((blockIdx.x == 0) ? WS_UNU : WS_UNV);
  float* Qout     = ws + ((blockIdx.x == 0) ? WS_QU  : WS_QV);
  int tid = threadIdx.x;

  for (int e = tid; e < HID * HID; e += 1024)
    Q[e] = ((e >> 8) == (e & 255)) ? 1.f : 0.f;
  __syncthreads();

  for (int i = HID - 1; i >= 0; --i) {
    if (tid < HID) u[tid] = Un[i * HID + tid];
    __syncthreads();
    int j = tid & 255, part = tid >> 8;
    float s = 0.f;
    for (int k = part * 64; k < part * 64 + 64; ++k)
      s += u[k] * Q[k * HID + j];
    t4[tid] = s;
    __syncthreads();
    if (tid < HID)
      t[tid] = t4[tid] + t4[tid + 256] + t4[tid + 512] + t4[tid + 768];
    __syncthreads();
    for (int e = tid; e < HID * HID; e += 1024)
      Q[e] -= 2.f * u[e >> 8] * t[e & 255];
    __syncthreads();
  }
  for (int e = tid; e < HID * HID; e += 1024) Qout[e] = Q[e];
}

// ---- W[i][j] = (1 - clamp(|alpha[j]|,0,1)) * V[j][i] -----------------------
__global__ void build_W_kernel(const float* __restrict__ alpha,
                               float* __restrict__ ws) {
  int i = blockIdx.x, j = threadIdx.x;
  float S = 1.f - fminf(fabsf(alpha[j]), 1.f);
  ws[WS_W + i * HID + j] = S * ws[WS_QV + j * HID + i];
}

// ---- generic fp32 WMMA GEMM, compile-time trans + NT tiles per wave -------
// Each wave produces NT adjacent 16x16 tiles (A operand reused NT times).
// grid.y = K-split index; each split writes its own M x N partial at
// C + blockIdx.y*M*ldc (grid.y==1 -> writes C directly).
// Wave-uniform guard only -> EXEC all ones around v_wmma (ISA 7.12 req).
template <int TRANSA, int TRANSB, int NT>
__global__ void gemm_wmma_kernel(const float* __restrict__ A,
                                 const float* __restrict__ B,
                                 float* __restrict__ C,
                                 int M, int N, int K,
                                 int lda, int ldb, int ldc, int kChunk) {
  int lane = threadIdx.x & 31;
  int wave = threadIdx.x >> 5;
  int ncg = N / (16 * NT);
  int grp = blockIdx.x * (blockDim.x >> 5) + wave;
  if (grp >= (M >> 4) * ncg) return;
  int tm  = (grp / ncg) << 4;
  int tn0 = (grp % ncg) * (16 * NT);
  int k0  = blockIdx.y * kChunk;

  int m   = lane & 15;           // row in A-tile / col offset in B-tile
  int kk0 = (lane >> 4) << 1;    // lanes 0-15 -> K {0,1}; 16-31 -> {2,3}

  v8f acc[NT];
#pragma unroll
  for (int i = 0; i < NT; ++i) acc[i] = (v8f){};

  for (int k = k0; k < k0 + kChunk; k += 4) {
    v2f a;
    if (!TRANSA) {
      const float* p = A + (size_t)(tm + m) * lda + k + kk0;
      a.x = p[0]; a.y = p[1];
      __builtin_prefetch(p + 64, 0, 3);
    } else {
      const float* p0 = A + (size_t)(k + kk0) * lda + tm + m;
      a.x = p0[0]; a.y = p0[lda];
      __builtin_prefetch(p0 + (size_t)64 * lda, 0, 3);
    }
#pragma unroll
    for (int i = 0; i < NT; ++i) {
      v2f b;
      if (!TRANSB) {
        const float* q = B + (size_t)(k + kk0) * ldb + tn0 + 16 * i + m;
        b.x = q[0]; b.y = q[ldb];
      } else {
        const float* q = B + (size_t)(tn0 + 16 * i + m) * ldb + k + kk0;
        b.x = q[0]; b.y = q[1];
      }
      acc[i] = __builtin_amdgcn_wmma_f32_16x16x4_f32(
          false, a, false, b, (short)0, acc[i], false, false);
    }
  }

  float* Cout = C + (size_t)blockIdx.y * M * ldc;
  int rbase = tm + ((lane >> 4) << 3);
  int col0  = tn0 + (lane & 15);
#pragma unroll
  for (int i = 0; i < NT; ++i)
    for (int r = 0; r < 8; ++r)
      Cout[(size_t)(rbase + r) * ldc + col0 + 16 * i] = acc[i][r];
}

// ---- O-recurrence GEMM: Odst(1024x256) = Osrc @ A, A staged in LDS --------
// Block = 256 thr = 8 waves covering 32 rows x 256 cols (NT=4 per wave).
// dyn LDS = 256KB (full A matrix), filled via async global->LDS when
// available (ASYNCcnt path), else float4 copy.
__global__ void gemm_olds_kernel(const float* __restrict__ Osrc,
                                 const float* __restrict__ Amat,
                                 float* __restrict__ Odst) {
  extern __shared__ float Blds[];          // 256x256
  int tid = threadIdx.x;
  stage_to_lds_256(Amat, Blds, HID * HID, tid);

  int lane = tid & 31, wave = tid >> 5;
  int tm  = blockIdx.x * 32 + (wave >> 2) * 16;
  int tn0 = (wave & 3) * 64;
  int m   = lane & 15;
  int kk0 = (lane >> 4) << 1;

  v8f acc[4];
#pragma unroll
  for (int i = 0; i < 4; ++i) acc[i] = (v8f){};

  for (int k = 0; k < HID; k += 4) {
    v2f a;
    const float* p = Osrc + (size_t)(tm + m) * HID + k + kk0;
    a.x = p[0]; a.y = p[1];
    const float* b0 = Blds + (k + kk0) * HID + tn0 + m;
    const float* b1 = b0 + HID;
#pragma unroll
    for (int i = 0; i < 4; ++i) {
      v2f b;
      b.x = b0[16 * i]; b.y = b1[16 * i];
      acc[i] = __builtin_amdgcn_wmma_f32_16x16x4_f32(
          false, a, false, b, (short)0, acc[i], false, false);
    }
  }

  int rbase = tm + ((lane >> 4) << 3);
  int col0  = tn0 + (lane & 15);
#pragma unroll
  for (int i = 0; i < 4; ++i)
    for (int r = 0; r < 8; ++r)
      Odst[(size_t)(rbase + r) * HID + col0 + 16 * i] = acc[i][r];
}

// ---- fixed-order split reduction: out[i] = sum_s in[s*stride + i] ---------
__global__ void reduce_splits_kernel(const float* __restrict__ in,
                                     float* __restrict__ out,
                                     int n, int splits, int stride) {
  int i = blockIdx.x * blockDim.x + threadIdx.x;
  if (i < n) {
    float s = 0.f;
    for (int p = 0; p < splits; ++p) s += in[(size_t)p * stride + i];
    out[i] = s;
  }
}

// ---- single-WGP Gauss-Jordan solve: G x = By, 32 RHS (dyn LDS 288KB) ------
__global__ void solve_kernel(float* __restrict__ ws) {
  extern __shared__ float lds[];
  float* Gs = lds;                  // 65536
  float* bs = lds + HID * HID;      // 8192
  __shared__ float pivinv;
  int tid = threadIdx.x;            // 256

  stage_to_lds_256(ws + WS_G, Gs, HID * HID, tid);
  for (int e = tid; e < HID * BATCH; e += HID) bs[e] = ws[WS_BY + e];
  __syncthreads();

  for (int p = 0; p < HID; ++p) {
    if (tid == 0) pivinv = 1.f / Gs[p * HID + p];
    __syncthreads();
    Gs[p * HID + tid] *= pivinv;
    if (tid < BATCH) bs[p * BATCH + tid] *= pivinv;
    __syncthreads();
    if (tid != p) {
      float f = Gs[tid * HID + p];
      for (int jj = p; jj < HID; ++jj)
        Gs[tid * HID + jj] -= f * Gs[p * HID + jj];
      for (int c = 0; c < BATCH; ++c)
        bs[tid * BATCH + c] -= f * bs[p * BATCH + c];
    }
    __syncthreads();
  }
  for (int e = tid; e < HID * BATCH; e += HID) ws[WS_X0 + e] = bs[e];
}

// ---- Yhat = O @ x0 fused with residual SSE, per-block partial -------------
// Wave covers 16 rows x 32 cols (NT=2). 3200 wave-groups -> 400 blocks.
__global__ void residual_loss_kernel(const float* __restrict__ y,
                                     float* __restrict__ ws) {
  __shared__ float red[256];
  int lane = threadIdx.x & 31;
  int wave = threadIdx.x >> 5;
  int grp = blockIdx.x * 8 + wave;
  bool active = grp < (SO >> 4);

  float part = 0.f;
  if (active) {
    int tm  = grp << 4;
    int m   = lane & 15;
    int kk0 = (lane >> 4) << 1;
    const float* O  = ws + WS_O;
    const float* x0 = ws + WS_X0;
    v8f acc[2];
#pragma unroll
    for (int i = 0; i < 2; ++i) acc[i] = (v8f){};
    for (int k = 0; k < HID; k += 4) {
      v2f a;
      const float* p = O + (size_t)(tm + m) * HID + k + kk0;
      a.x = p[0]; a.y = p[1];
#pragma unroll
      for (int i = 0; i < 2; ++i) {
        v2f b;
        const float* q = x0 + (size_t)(k + kk0) * BATCH + 16 * i + m;
        b.x = q[0]; b.y = q[BATCH];
        acc[i] = __builtin_amdgcn_wmma_f32_16x16x4_f32(
            false, a, false, b, (short)0, acc[i], false, false);
      }
    }
    int rbase = tm + ((lane >> 4) << 3);
#pragma unroll
    for (int i = 0; i < 2; ++i) {
      int col = 16 * i + (lane & 15);
      for (int r = 0; r < 8; ++r) {
        float w = y[(size_t)col * SO + rbase + r] - acc[i][r];
        part += w * w;
      }
    }
  }
  red[threadIdx.x] = part;
  __syncthreads();
  for (int s = 128; s > 0; s >>= 1) {
    if (threadIdx.x < s) red[threadIdx.x] += red[threadIdx.x + s];
    __syncthreads();
  }
  if (threadIdx.x == 0) ws[WS_RES + blockIdx.x] = red[0];
}

// ---- finalize: sum residual partials + logdet term, write loss ------------
__global__ void finalize_kernel(const float* __restrict__ logdet,
                                const float* __restrict__ ws,
                                float* __restrict__ out) {
  __shared__ float red[256];
  float s = 0.f;
  for (int i = threadIdx.x; i < RES_BLOCKS; i += 256) s += ws[WS_RES + i];
  float ld = 0.f;
  for (int i = threadIdx.x; i < BATCH * SEQ; i += 256) {
    float v = logdet[i] * (1.f / (float)OBS);
    ld += v * v;
  }
  red[threadIdx.x] = s;
  __syncthreads();
  for (int st = 128; st > 0; st >>= 1) {
    if (threadIdx.x < st) red[threadIdx.x] += red[threadIdx.x + st];
    __syncthreads();
  }
  float sse = red[0];
  __syncthreads();
  red[threadIdx.x] = ld;
  __syncthreads();
  for (int st = 128; st > 0; st >>= 1) {
    if (threadIdx.x < st) red[threadIdx.x] += red[threadIdx.x + st];
    __syncthreads();
  }
  if (threadIdx.x == 0)
    out[0] = 0.5f * sse / ((float)SO * (float)BATCH)
           + red[0] / (float)(BATCH * SEQ);
}

// ---------------------------------------------------------------------------
extern "C" void kernel_launch(void* const* d_in, const int* in_sizes, int n_in,
                              void* d_out, int out_size, void* d_ws, size_t ws_size,
                              hipStream_t stream) {
  const float* y       = (const float*)d_in[0];
  const float* logdet  = (const float*)d_in[1];
  const float* U_param = (const float*)d_in[2];
  const float* V_param = (const float*)d_in[3];
  const float* alpha   = (const float*)d_in[4];
  const float* Cmat    = (const float*)d_in[5];
  float* out = (float*)d_out;
  float* ws  = (float*)d_ws;
  (void)in_sizes; (void)n_in; (void)out_size; (void)ws_size;

  // 1) normalize Householder vectors
  normalize_rows_kernel<<<dim3(HID, 2), HID, 0, stream>>>(U_param, V_param, ws);

  // 2) build unitary U, V (one WGP each, Q in LDS)
  size_t hh_lds = (size_t)(HID * HID + HID + HID + 1024) * sizeof(float);
  householder_chain_kernel<<<2, 1024, hh_lds, stream>>>(ws);

  // 3) W = S .* V^T ; A = Q_U @ W   (M=N=K=256, NT=4 -> 64 waves, 8 blocks)
  build_W_kernel<<<HID, HID, 0, stream>>>(alpha, ws);
  gemm_wmma_kernel<0, 0, 4><<<dim3(8, 1), 256, 0, stream>>>(
      ws + WS_QU, ws + WS_W, ws + WS_A,
      HID, HID, HID, HID, HID, HID, HID);

  // 4) O[0]=C ; O[t]=O[t-1]@A with A staged in LDS (32 blocks, 256KB LDS)
  (void)hipMemcpyAsync(ws + WS_O, Cmat, (size_t)OBS * HID * sizeof(float),
                       hipMemcpyDeviceToDevice, stream);
  size_t a_lds = (size_t)HID * HID * sizeof(float);
  for (int t = 1; t < SEQ; ++t) {
    gemm_olds_kernel<<<OBS / 32, 256, a_lds, stream>>>(
        ws + WS_O + (size_t)(t - 1) * OBS * HID,
        ws + WS_A,
        ws + WS_O + (size_t)t * OBS * HID);
  }

  // 5) G = O^T O (split-K partials) ; By = O^T Y^T (split-K partials)
  gemm_wmma_kernel<1, 0, 4><<<dim3(8, KSPLIT), 256, 0, stream>>>(
      ws + WS_O, ws + WS_O, ws + WS_GP,
      HID, HID, SO, HID, HID, HID, KCHUNK);
  reduce_splits_kernel<<<(HID * HID + 255) / 256, 256, 0, stream>>>(
      ws + WS_GP, ws + WS_G, HID * HID, KSPLIT, HID * HID);

  gemm_wmma_kernel<1, 1, 2><<<dim3(2, KSPLIT), 256, 0, stream>>>(
      ws + WS_O, y, ws + WS_BYP,
      HID, BATCH, SO, HID, SO, BATCH, KCHUNK);
  reduce_splits_kernel<<<(HID * BATCH + 255) / 256, 256, 0, stream>>>(
      ws + WS_BYP, ws + WS_BY, HID * BATCH, KSPLIT, HID * BATCH);

  // 6) solve G x0 = By (Gauss-Jordan in LDS)
  size_t sv_lds = (size_t)(HID * HID + HID * BATCH) * sizeof(float);
  solve_kernel<<<1, HID, sv_lds, stream>>>(ws);

  // 7) fused Yhat + residual partials; finalize with logdet term
  residual_loss_kernel<<<RES_BLOCKS, 256, 0, stream>>>(y, ws);
  finalize_kernel<<<1, 256, 0, stream>>>(logdet, ws, out);
}